// MyLoss_38027640439091
// MI455X (gfx1250) — compile-verified
//
#include <hip/hip_runtime.h>
#include <stdint.h>

// ---------------------------------------------------------------------------
// Weighted cross-entropy (sum of -w[t] * log_softmax(logits)[row, t]).
// N = 8192 rows, C = 50257 classes, fp32.
// Memory-bound: 1.65 GB logits @ 23.3 TB/s => ~71 us floor. One pass over
// memory via chunked online softmax; data staged with CDNA5 async
// global->LDS copies (ASYNCcnt) in a per-thread double-buffered pipeline.
// Exp work ~412M v_exp_f32 -- well under the HBM floor.
// ---------------------------------------------------------------------------

#define L2E 1.4426950408889634f   // log2(e)
#define LN2 0.6931471805599453f   // ln(2)

constexpr int kC      = 50257;
constexpr int kVec4   = kC / 4;              // 12564 full float4s (covers 50256)
constexpr int kBlock  = 256;
constexpr int kJ      = 4;                   // b128 loads per thread per chunk
constexpr int kChunkV = kBlock * kJ;         // 1024 float4 per chunk
constexpr int kChunks = (kVec4 + kChunkV - 1) / kChunkV;  // 13

// Async global->LDS copy, GVS mode: mem = SADDR64 + VADDR32. VDST VGPR holds
// the wave-relative LDS byte address (low 32 bits of the flat LDS pointer).
__device__ __forceinline__ void async_copy_b128(const float4* lds_dst,
                                                const float*  srow,
                                                uint32_t      byte_off) {
  uint32_t lds_off = (uint32_t)(uintptr_t)lds_dst;
  asm volatile("global_load_async_to_lds_b128 %0, %1, %2"
               :: "v"(lds_off), "v"(byte_off), "s"(srow)
               : "memory");
}

// Issue one chunk: EXACTLY 4 async instructions per wave, unconditionally.
// Out-of-range lanes clamp their address to a valid in-bounds float4 so the
// per-wave ASYNCcnt bookkeeping (4 per chunk) is uniform; the junk data is
// masked out at compute time. No divergent branches around the asm.
__device__ __forceinline__ void issue_chunk(int c, float4* buf,
                                            const float* rowbase, int tid) {
  const int stage = c & 1;
#pragma unroll
  for (int j = 0; j < kJ; ++j) {
    int v = c * kChunkV + j * kBlock + tid;            // float4 index in row
    v = (v < kVec4) ? v : (kVec4 - 1);                 // clamp (v_min)
    async_copy_b128(buf + (stage * kChunkV + j * kBlock + tid),
                    rowbase, (uint32_t)v * 16u);
  }
}

// Merge two online-softmax states: (m,s) <- (m,s) (+) (om,os)
__device__ __forceinline__ void merge(float& m, float& s, float om, float os) {
  const float nm = fmaxf(m, om);
  s = s  * __builtin_amdgcn_exp2f((m  - nm) * L2E)
    + os * __builtin_amdgcn_exp2f((om - nm) * L2E);
  m = nm;
}

__device__ __forceinline__ float vmax4(float4 v) {
  return fmaxf(fmaxf(v.x, v.y), fmaxf(v.z, v.w));
}

__global__ __launch_bounds__(kBlock)
void ce_row_kernel(const float* __restrict__ logits,
                   const int*   __restrict__ target,
                   const float* __restrict__ weights,
                   float*       __restrict__ rowloss) {
  __shared__ float4 buf[2 * kChunkV];          // 32 KB double buffer
  __shared__ float  wm[kBlock / 32], ws[kBlock / 32];

  const int tid = threadIdx.x;
  const int row = blockIdx.x;
  const float* rowbase = logits + (size_t)row * kC;

  // Prologue: two chunks in flight (kChunks == 13 >= 2). 8 async instrs out.
  issue_chunk(0, buf, rowbase, tid);
  issue_chunk(1, buf, rowbase, tid);

  float m = -__builtin_inff();
  float s = 0.0f;

  for (int c = 0; c < kChunks; ++c) {
    // Outstanding: chunk c (4 instr) and maybe c+1 (4 instr). Every chunk is
    // exactly 4 instructions per wave and async loads complete in order, so
    // asynccnt<=4 guarantees chunk c has fully landed in LDS.
    if (c + 1 < kChunks) {
      asm volatile("s_wait_asynccnt 4" ::: "memory");
    } else {
      asm volatile("s_wait_asynccnt 0" ::: "memory");
    }

    const int stage = c & 1;
    float4 x[kJ];
#pragma unroll
    for (int j = 0; j < kJ; ++j) {
      x[j] = buf[stage * kChunkV + j * kBlock + tid];   // ds_load_b128
      const int v = c * kChunkV + j * kBlock + tid;
      if (v >= kVec4) {                                 // tail-chunk mask
        x[j].x = x[j].y = x[j].z = x[j].w = -__builtin_inff();
      }
    }

    // Local max over 16 values, then one exp per element vs max(running max, local max),
    // single rescale to merge with the running state (~1.06 exps/element).
    float m16 = fmaxf(fmaxf(vmax4(x[0]), vmax4(x[1])),
                      fmaxf(vmax4(x[2]), vmax4(x[3])));
    const float nm = fmaxf(m, m16);
    float acc = 0.0f;
#pragma unroll
    for (int j = 0; j < kJ; ++j) {
      acc += __builtin_amdgcn_exp2f((x[j].x - nm) * L2E);
      acc += __builtin_amdgcn_exp2f((x[j].y - nm) * L2E);
      acc += __builtin_amdgcn_exp2f((x[j].z - nm) * L2E);
      acc += __builtin_amdgcn_exp2f((x[j].w - nm) * L2E);
    }
    s = s * __builtin_amdgcn_exp2f((m - nm) * L2E) + acc;
    m = nm;

    // Refill the slot we just consumed. Ensure the ds reads above have
    // completed before the async engine may overwrite the slot.
    if (c + 2 < kChunks) {
      asm volatile("s_wait_dscnt 0" ::: "memory");
      issue_chunk(c + 2, buf, rowbase, tid);
    }
  }

  // Wave32 reduction of (m, s).
#pragma unroll
  for (int off = 16; off > 0; off >>= 1) {
    const float om = __shfl_xor(m, off, 32);
    const float os = __shfl_xor(s, off, 32);
    merge(m, s, om, os);
  }
  const int wave = tid >> 5, lane = tid & 31;
  if (lane == 0) { wm[wave] = m; ws[wave] = s; }
  __syncthreads();

  if (tid == 0) {
    float M = wm[0], S = ws[0];
#pragma unroll
    for (int i = 1; i < kBlock / 32; ++i) merge(M, S, wm[i], ws[i]);
    // Tail element (index C-1 = 50256) as a 1-element state (m = x, s = 1).
    merge(M, S, rowbase[kC - 1], 1.0f);

    const int   t   = target[row];
    const float lse = M + LN2 * __builtin_amdgcn_logf(S);   // v_log_f32 = log2
    rowloss[row] = -weights[t] * (rowbase[t] - lse);
  }
}

// Deterministic final reduction: single block, LDS tree.
__global__ __launch_bounds__(256)
void sum_kernel(const float* __restrict__ in, float* __restrict__ out, int n) {
  __shared__ float sm[256];
  float acc = 0.0f;
  for (int i = threadIdx.x; i < n; i += 256) acc += in[i];
  sm[threadIdx.x] = acc;
  __syncthreads();
#pragma unroll
  for (int off = 128; off > 0; off >>= 1) {
    if ((int)threadIdx.x < off) sm[threadIdx.x] += sm[threadIdx.x + off];
    __syncthreads();
  }
  if (threadIdx.x == 0) out[0] = sm[0];
}

extern "C" void kernel_launch(void* const* d_in, const int* in_sizes, int n_in,
                              void* d_out, int out_size, void* d_ws, size_t ws_size,
                              hipStream_t stream) {
  const float* logits  = (const float*)d_in[0];
  const int*   target  = (const int*)d_in[1];
  const float* weights = (const float*)d_in[2];
  const int N = in_sizes[1];                 // 8192 rows (target count)

  float* rowloss = (float*)d_ws;             // N floats of scratch

  ce_row_kernel<<<N, kBlock, 0, stream>>>(logits, target, weights, rowloss);
  sum_kernel<<<1, 256, 0, stream>>>(rowloss, (float*)d_out, N);
}